// CamCenterLoss_15710990369513
// MI455X (gfx1250) — compile-verified
//
#include <hip/hip_runtime.h>
#include <hip/hip_bf16.h>

// Problem constants (from reference)
#define NN 16384
#define DD 2048
#define SS 8192          // NUM_LABELS * NUM_CAMS = 1024 * 8
#define TPB 256
#define PER_T 8          // DD / TPB columns owned per thread
#define CACHE_ROWS 6     // rows of a segment staged in LDS (48 KB)

typedef float v2f __attribute__((ext_vector_type(2)));
typedef float v8f __attribute__((ext_vector_type(8)));

__global__ void k_zero(float* out, int* counts) {
    int i = blockIdx.x * blockDim.x + threadIdx.x;
    if (i < SS) counts[i] = 0;
    if (i == 0) out[0] = 0.0f;
}

__global__ void k_seg_count(const int* __restrict__ labels,
                            const int* __restrict__ cams,
                            int* __restrict__ seg,
                            int* __restrict__ counts, int n) {
    int i = blockIdx.x * blockDim.x + threadIdx.x;
    if (i < n) {
        int s = labels[i] * 8 + cams[i];
        seg[i] = s;
        atomicAdd(&counts[s], 1);
    }
}

// One block: exclusive prefix sum of counts[SS] -> offsets, cursor
__global__ void k_scan(const int* __restrict__ counts,
                       int* __restrict__ offsets,
                       int* __restrict__ cursor) {
    __shared__ int tsum[TPB];
    const int per = SS / TPB;  // 32
    int tid = threadIdx.x;
    int base = tid * per;
    int run = 0;
    #pragma unroll
    for (int k = 0; k < per; ++k) run += counts[base + k];
    tsum[tid] = run;
    __syncthreads();
    if (tid == 0) {
        int acc = 0;
        for (int i = 0; i < TPB; ++i) { int t = tsum[i]; tsum[i] = acc; acc += t; }
    }
    __syncthreads();
    int acc = tsum[tid];
    for (int k = 0; k < per; ++k) {
        offsets[base + k] = acc;
        cursor[base + k]  = acc;
        acc += counts[base + k];
    }
}

__global__ void k_scatter(const int* __restrict__ seg,
                          int* __restrict__ cursor,
                          int* __restrict__ rowlist, int n) {
    int i = blockIdx.x * blockDim.x + threadIdx.x;
    if (i < n) {
        int pos = atomicAdd(&cursor[seg[i]], 1);
        rowlist[pos] = i;
    }
}

// One block per segment: async-stage member rows in LDS, compute mean,
// then SmoothL1 against the mean; WMMA-based wave reduction; atomic scalar out.
__global__ void __launch_bounds__(TPB) k_segloss(const float* __restrict__ feats,
                                                 const int* __restrict__ rowlist,
                                                 const int* __restrict__ offsets,
                                                 const int* __restrict__ counts,
                                                 float* __restrict__ out,
                                                 float scale) {
    __shared__ __align__(16) float lds[CACHE_ROWS * DD];
    __shared__ float wsum[TPB / 32];

    int s = blockIdx.x;
    int cnt = counts[s];
    if (cnt == 0) return;                 // block-uniform: safe before barriers
    int base = offsets[s];
    int tid = threadIdx.x;
    int ncache = cnt < CACHE_ROWS ? cnt : CACHE_ROWS;

    // --- Stage up to CACHE_ROWS rows into LDS via CDNA5 async copy ---
    for (int r = 0; r < ncache; ++r) {
        int row = rowlist[base + r];
        const float* src = feats + (size_t)row * DD;
        unsigned lbase = (unsigned)(uintptr_t)(&lds[r * DD]);
        #pragma unroll
        for (int q = 0; q < 2; ++q) {
            int chunk = tid + q * TPB;                 // 512 x 16B chunks per row
            unsigned loff = lbase + (unsigned)chunk * 16u;
            const void* g = (const void*)(src + chunk * 4);
            asm volatile("global_load_async_to_lds_b128 %0, %1, off"
                         :: "v"(loff), "v"(g) : "memory");
        }
    }
    asm volatile("s_wait_asynccnt 0" ::: "memory");
    __syncthreads();

    // --- Segment mean over owned columns: col(j) = tid + j*TPB (coalesced) ---
    float m[PER_T];
    #pragma unroll
    for (int j = 0; j < PER_T; ++j) m[j] = 0.0f;
    for (int r = 0; r < cnt; ++r) {
        if (r < ncache) {
            #pragma unroll
            for (int j = 0; j < PER_T; ++j) m[j] += lds[r * DD + j * TPB + tid];
        } else {
            const float* src = feats + (size_t)rowlist[base + r] * DD;
            #pragma unroll
            for (int j = 0; j < PER_T; ++j) m[j] += src[j * TPB + tid];
        }
    }
    float inv = 1.0f / (float)cnt;
    #pragma unroll
    for (int j = 0; j < PER_T; ++j) m[j] *= inv;

    // --- SmoothL1 (beta=1) vs mean ---
    float a = 0.0f;
    for (int r = 0; r < cnt; ++r) {
        const float* src = (r >= ncache) ? (feats + (size_t)rowlist[base + r] * DD)
                                         : nullptr;
        #pragma unroll
        for (int j = 0; j < PER_T; ++j) {
            float v = (r < ncache) ? lds[r * DD + j * TPB + tid] : src[j * TPB + tid];
            float d = v - m[j];
            float ad = fabsf(d);
            a += (ad < 1.0f) ? (0.5f * d * d) : (ad - 0.5f);
        }
    }

    // --- Wave32 reduction via V_WMMA_F32_16X16X4_F32 ---
    // A layout (16x4 f32): lane L<16 -> A[L][0]=a (VGPR0), A[L][1]=0 (VGPR1);
    // lane L>=16 -> A[L-16][2]=a, A[L-16][3]=0. B = ones(4x16).
    // => D[m][n] = a_m + a_(m+16). Per-lane sum of the 8 D VGPRs gives the
    // low-half total (lanes 0-15) / high-half total (lanes 16-31).
    v2f A; A[0] = a;    A[1] = 0.0f;
    v2f B; B[0] = 1.0f; B[1] = 1.0f;
    v8f C = {};
    v8f Dm = __builtin_amdgcn_wmma_f32_16x16x4_f32(
        /*neg_a=*/false, A, /*neg_b=*/false, B,
        /*c_mod=*/(short)0, C, /*reuse_a=*/false, /*reuse_b=*/false);
    float sv = Dm[0] + Dm[1] + Dm[2] + Dm[3] + Dm[4] + Dm[5] + Dm[6] + Dm[7];
    sv += __shfl_xor(sv, 16, 32);   // combine halves -> full wave sum (all lanes)

    int wid = tid >> 5;
    if ((tid & 31) == 0) wsum[wid] = sv;
    __syncthreads();
    if (tid == 0) {
        float t = 0.0f;
        #pragma unroll
        for (int w = 0; w < TPB / 32; ++w) t += wsum[w];
        atomicAdd(out, t * scale);
    }
}

extern "C" void kernel_launch(void* const* d_in, const int* in_sizes, int n_in,
                              void* d_out, int out_size, void* d_ws, size_t ws_size,
                              hipStream_t stream) {
    const float* feats = (const float*)d_in[0];
    const int* labels  = (const int*)d_in[1];
    const int* cams    = (const int*)d_in[2];
    float* out = (float*)d_out;
    int n = in_sizes[1];                   // 16384 rows

    // Workspace layout (all int32): seg[N] | counts[S] | offsets[S] | cursor[S] | rowlist[N]
    int* seg     = (int*)d_ws;
    int* counts  = seg + n;
    int* offsets = counts + SS;
    int* cursor  = offsets + SS;
    int* rowlist = cursor + SS;

    k_zero<<<(SS + TPB - 1) / TPB, TPB, 0, stream>>>(out, counts);
    k_seg_count<<<(n + TPB - 1) / TPB, TPB, 0, stream>>>(labels, cams, seg, counts, n);
    k_scan<<<1, TPB, 0, stream>>>(counts, offsets, cursor);
    k_scatter<<<(n + TPB - 1) / TPB, TPB, 0, stream>>>(seg, cursor, rowlist, n);

    float scale = 1.0f / ((float)n * (float)DD);
    k_segloss<<<SS, TPB, 0, stream>>>(feats, rowlist, offsets, counts, out, scale);
}